// SGGCN_69002944578214
// MI455X (gfx1250) — compile-verified
//
#include <hip/hip_runtime.h>
#include <cstdint>

typedef float v2f __attribute__((ext_vector_type(2)));
typedef float v8f __attribute__((ext_vector_type(8)));

#define FEAT 128

// ---------------------------------------------------------------------------
// Edge propagate: one wave (32 lanes) per edge. Lane l gathers float4 at
// feature offset l*4 (coalesced 512B per wave) and atomically accumulates
// into the destination row. src/dst are int64 per the reference.
// ---------------------------------------------------------------------------
__global__ __launch_bounds__(256) void propagate_kernel(
    const float* __restrict__ x,
    const long long* __restrict__ srcI,
    const long long* __restrict__ dstI,
    const float* __restrict__ ew,
    float* __restrict__ out,
    int n_edges) {
  int gid  = blockIdx.x * blockDim.x + threadIdx.x;
  int edge = gid >> 5;
  int lane = gid & 31;
  if (edge >= n_edges) return;
  long long s = srcI[edge];
  long long d = dstI[edge];
  float w = ew[edge];
  const float4* xs = reinterpret_cast<const float4*>(x + (size_t)s * FEAT);
  float4 v = xs[lane];
  float* o = out + (size_t)d * FEAT + lane * 4;
  atomicAdd(o + 0, w * v.x);
  atomicAdd(o + 1, w * v.y);
  atomicAdd(o + 2, w * v.z);
  atomicAdd(o + 3, w * v.w);
}

// ---------------------------------------------------------------------------
// FP32 WMMA GEMM: D[N, NT*16] = A[N,128] @ W[NT*16,128]^T + bias (opt ReLU).
// One wave computes a 16-row stripe across all NT column tiles using
// V_WMMA_F32_16X16X4_F32, reusing the A fragment across the N-tile loop.
// A-frag: lane m=lane&15, kh=2*(lane>>4): {A[m][kb], A[m][kb+1]}
// B-frag (B = W^T): symmetric: {W[n][kb], W[n][kb+1]}, n = tile*16 + (lane&15)
// C/D: VGPR v -> row v + 8*(lane>=16), col lane&15.
// ---------------------------------------------------------------------------
template <int NT>
__global__ __launch_bounds__(256) void gemm_kernel(
    const float* __restrict__ A,
    const float* __restrict__ W,
    const float* __restrict__ bias,
    float* __restrict__ D,
    int nrows, int do_relu) {
  const int NCOLS = NT * 16;
  int wave = (blockIdx.x * blockDim.x + threadIdx.x) >> 5;
  int lane = threadIdx.x & 31;
  int row0 = wave * 16;
  if (row0 >= nrows) return;  // wave-uniform exit: live waves keep EXEC all-1s

  const int m  = lane & 15;
  const int kh = (lane >> 4) << 1;  // 0 or 2

  v8f acc[NT] = {};
  const float* arow = A + (size_t)(row0 + m) * FEAT;
  const float* wrow = W + (size_t)m * FEAT;

  for (int kk = 0; kk < 32; ++kk) {
    const int kb = kk * 4 + kh;
    v2f a = *reinterpret_cast<const v2f*>(arow + kb);
#pragma unroll
    for (int t = 0; t < NT; ++t) {
      v2f b = *reinterpret_cast<const v2f*>(wrow + t * 16 * FEAT + kb);
      acc[t] = __builtin_amdgcn_wmma_f32_16x16x4_f32(
          /*neg_a=*/false, a, /*neg_b=*/false, b,
          /*c_mod=*/(short)0, acc[t], /*reuse_a=*/false, /*reuse_b=*/false);
    }
  }

  const int rbase = row0 + ((lane >> 4) << 3);  // +8 for upper half-lanes
#pragma unroll
  for (int t = 0; t < NT; ++t) {
    const int col = t * 16 + m;
    const float bv = bias[col];
#pragma unroll
    for (int v = 0; v < 8; ++v) {
      float val = acc[t][v] + bv;
      if (do_relu) val = fmaxf(val, 0.0f);
      D[(size_t)(rbase + v) * NCOLS + col] = val;
    }
  }
}

// ---------------------------------------------------------------------------
// Column statistics: per-column sum and sum-of-squares over all rows.
// Block of 128 threads reads whole rows (coalesced); grid-strided over rows;
// 256 float atomics per block into stats[0..255].
// ---------------------------------------------------------------------------
__global__ void colstats_kernel(const float* __restrict__ A,
                                float* __restrict__ stats, int nrows) {
  int col = threadIdx.x;  // blockDim.x == 128
  float s = 0.0f, s2 = 0.0f;
  for (int r = blockIdx.x; r < nrows; r += gridDim.x) {
    float v = A[(size_t)r * FEAT + col];
    s  += v;
    s2 += v * v;
  }
  atomicAdd(&stats[col], s);
  atomicAdd(&stats[FEAT + col], s2);
}

// ---------------------------------------------------------------------------
// Fused BatchNorm (training-mode batch stats, biased var) + ReLU.
// ---------------------------------------------------------------------------
__global__ __launch_bounds__(256) void bnrelu_kernel(
    const float* __restrict__ t, const float* __restrict__ stats,
    const float* __restrict__ g, const float* __restrict__ be,
    float* __restrict__ out, int nrows) {
  long long total = (long long)nrows * FEAT;
  long long i = (long long)blockIdx.x * blockDim.x + threadIdx.x;
  if (i >= total) return;
  int col = (int)(i & (FEAT - 1));
  float invN  = 1.0f / (float)nrows;
  float mean  = stats[col] * invN;
  float var   = stats[FEAT + col] * invN - mean * mean;
  float inv   = rsqrtf(var + 1e-5f);
  float v = (t[i] - mean) * inv * g[col] + be[col];
  out[i] = fmaxf(v, 0.0f);
}

extern "C" void kernel_launch(void* const* d_in, const int* in_sizes, int n_in,
                              void* d_out, int out_size, void* d_ws, size_t ws_size,
                              hipStream_t stream) {
  const float*     x   = (const float*)d_in[0];
  const long long* ei  = (const long long*)d_in[1];
  const float*     ew  = (const float*)d_in[2];
  const float *W1 = (const float*)d_in[3],  *b1  = (const float*)d_in[4];
  const float *W2 = (const float*)d_in[5],  *b2  = (const float*)d_in[6];
  const float *W3 = (const float*)d_in[7],  *b3  = (const float*)d_in[8];
  const float *g1 = (const float*)d_in[9],  *be1 = (const float*)d_in[10];
  const float *g2 = (const float*)d_in[11], *be2 = (const float*)d_in[12];
  const float *pW1 = (const float*)d_in[13], *pb1 = (const float*)d_in[14];
  const float *pW2 = (const float*)d_in[15], *pb2 = (const float*)d_in[16];

  const int nrows  = in_sizes[0] / FEAT;   // N = 100000 (multiple of 16)
  const int nedges = in_sizes[2];          // E
  const long long* srcI = ei;
  const long long* dstI = ei + nedges;

  float* out    = (float*)d_out;
  float* logits = out;                               // [N, 64]
  float* hout   = out + (size_t)nrows * 64;          // [N, 128]
  float* zout   = hout + (size_t)nrows * FEAT;       // [N, 128]

  float* bufA  = (float*)d_ws;                       // propagate target
  float* bufB  = bufA + (size_t)nrows * FEAT;        // pre-BN GEMM target
  float* bufC  = bufB + (size_t)nrows * FEAT;        // h1 / z1
  float* stats = bufC + (size_t)nrows * FEAT;        // 256 floats
  const size_t bigBytes = (size_t)nrows * FEAT * sizeof(float);

  const dim3 blk(256);
  const int propBlocks = (nedges * 32 + 255) / 256;
  const int waves      = (nrows + 15) / 16;
  const int gemmBlocks = (waves * 32 + 255) / 256;
  const int bnBlocks   = (int)(((long long)nrows * FEAT + 255) / 256);

  // ---- layer 1: p1 = prop(x); t1 = p1@W1^T+b1; h1 = relu(bn(t1)) ----
  hipMemsetAsync(bufA, 0, bigBytes, stream);
  propagate_kernel<<<propBlocks, blk, 0, stream>>>(x, srcI, dstI, ew, bufA, nedges);
  gemm_kernel<8><<<gemmBlocks, blk, 0, stream>>>(bufA, W1, b1, bufB, nrows, 0);
  hipMemsetAsync(stats, 0, 2 * FEAT * sizeof(float), stream);
  colstats_kernel<<<1024, 128, 0, stream>>>(bufB, stats, nrows);
  bnrelu_kernel<<<bnBlocks, blk, 0, stream>>>(bufB, stats, g1, be1, bufC, nrows);

  // ---- layer 2: p2 = prop(h1); t2 = p2@W2^T+b2; h = relu(bn(t2)) ----
  hipMemsetAsync(bufA, 0, bigBytes, stream);
  propagate_kernel<<<propBlocks, blk, 0, stream>>>(bufC, srcI, dstI, ew, bufA, nedges);
  gemm_kernel<8><<<gemmBlocks, blk, 0, stream>>>(bufA, W2, b2, bufB, nrows, 0);
  hipMemsetAsync(stats, 0, 2 * FEAT * sizeof(float), stream);
  colstats_kernel<<<1024, 128, 0, stream>>>(bufB, stats, nrows);
  bnrelu_kernel<<<bnBlocks, blk, 0, stream>>>(bufB, stats, g2, be2, hout, nrows);

  // ---- logits = prop(h) @ W3^T + b3 ----
  hipMemsetAsync(bufA, 0, bigBytes, stream);
  propagate_kernel<<<propBlocks, blk, 0, stream>>>(hout, srcI, dstI, ew, bufA, nedges);
  gemm_kernel<4><<<gemmBlocks, blk, 0, stream>>>(bufA, W3, b3, logits, nrows, 0);

  // ---- z = relu(h@pW1^T+pb1) @ pW2^T + pb2 ----
  gemm_kernel<8><<<gemmBlocks, blk, 0, stream>>>(hout, pW1, pb1, bufC, nrows, 1);
  gemm_kernel<8><<<gemmBlocks, blk, 0, stream>>>(bufC, pW2, pb2, zout, nrows, 0);
}